// LTCCell_8581344658156
// MI455X (gfx1250) — compile-verified
//
#include <hip/hip_runtime.h>

// ---------------------------------------------------------------------------
// LTC cell for MI455X (gfx1250, wave32).
// Compute-bound on VALU transcendentals (~872M sigmoids). Strategy:
//  - sigmoid via v_tanh_f32 (1 transcendental) when available
//  - precompute -(mu*sigma) and W*erev once
//  - stage v_pre tiles in LDS via gfx1250 async global->LDS copies,
//    double-buffered, completion via s_wait_asynccnt
//  - params read per-j, reused across 8 batch rows per thread
// ---------------------------------------------------------------------------

#define BB   512   // batch
#define II   256   // sensory input dim
#define HH   512   // hidden dim
#define UNFOLDS 6

#define TILE_H 64
#define TILE_B 32
#define JT     64          // j-chunk staged in LDS
#define BLOCK  256         // 8 wave32
#define BT     8           // batch rows per thread (TILE_B / 4)

// ---------------- CDNA5 feature guards -------------------------------------
#if defined(__AMDGCN__) && __has_builtin(__builtin_amdgcn_global_load_async_to_lds_b128)
#define HAVE_ASYNC_LDS 1
#endif

#if defined(HAVE_ASYNC_LDS)
typedef int v4i_ltc __attribute__((vector_size(4 * sizeof(int))));
typedef __attribute__((address_space(1))) v4i_ltc* gvec_ptr;
typedef __attribute__((address_space(3))) v4i_ltc* lvec_ptr;
#endif

__device__ __forceinline__ void async_copy16(const float* g, float* l) {
#if defined(HAVE_ASYNC_LDS)
  __builtin_amdgcn_global_load_async_to_lds_b128(
      (gvec_ptr)g, (lvec_ptr)l, /*offset=*/0, /*cpol=*/0);
#else
  *(float4*)l = *(const float4*)g;
#endif
}

__device__ __forceinline__ void async_wait_le2() {
#if defined(HAVE_ASYNC_LDS)
#if __has_builtin(__builtin_amdgcn_s_wait_asynccnt)
  __builtin_amdgcn_s_wait_asynccnt(2);
#else
  asm volatile("s_wait_asynccnt 2" ::: "memory");
#endif
#endif
}

__device__ __forceinline__ void async_wait_le0() {
#if defined(HAVE_ASYNC_LDS)
#if __has_builtin(__builtin_amdgcn_s_wait_asynccnt)
  __builtin_amdgcn_s_wait_asynccnt(0);
#else
  asm volatile("s_wait_asynccnt 0" ::: "memory");
#endif
#endif
}

// tanh-based sigmoid: one transcendental on gfx1250 (v_tanh_f32)
__device__ __forceinline__ float fast_tanh(float x) {
#if defined(__AMDGCN__) && __has_builtin(__builtin_amdgcn_tanhf)
  return __builtin_amdgcn_tanhf(x);
#elif defined(__AMDGCN__)
  // tanh(x) = 2/(1+exp(-2x)) - 1 ; exp(-2x) = exp2(-2*log2(e)*x)
  float e = __builtin_amdgcn_exp2f(-2.8853900817779268f * x);
  return __builtin_fmaf(2.0f, __builtin_amdgcn_rcpf(1.0f + e), -1.0f);
#else
  return tanhf(x);
#endif
}

// ---------------- prep: nmsig = -(mu*sigma), wer = W*erev -------------------
__global__ void ltc_prep_kernel(const float* __restrict__ mu,
                                const float* __restrict__ sigma,
                                const float* __restrict__ W,
                                const float* __restrict__ erev,
                                float* __restrict__ nmsig,
                                float* __restrict__ wer, int n) {
  int i = blockIdx.x * blockDim.x + threadIdx.x;
  if (i < n) {
    nmsig[i] = -(mu[i] * sigma[i]);
    wer[i]   = W[i] * erev[i];
  }
}

// ---------------- sensory currents (once) -----------------------------------
__global__ __launch_bounds__(BLOCK) void ltc_sensory_kernel(
    const float* __restrict__ x,      // B x I
    const float* __restrict__ smu,    // I x H
    const float* __restrict__ ssig,   // I x H
    const float* __restrict__ sW,     // I x H
    const float* __restrict__ serev,  // I x H
    float* __restrict__ snum,         // B x H
    float* __restrict__ sden) {       // B x H
  __shared__ float xt[2][TILE_B * JT];

  const int tid = threadIdx.x;
  const int hl  = tid & 63;
  const int bs  = tid >> 6;  // 0..3
  const int h   = blockIdx.x * TILE_H + hl;
  const int b0  = blockIdx.y * TILE_B;

  float accn[BT], accd[BT];
#pragma unroll
  for (int k = 0; k < BT; ++k) { accn[k] = 0.f; accd[k] = 0.f; }

  // issue chunk 0
#pragma unroll
  for (int r = 0; r < 2; ++r) {
    int seg = tid + BLOCK * r;
    int bb = seg >> 4, js = seg & 15;
    async_copy16(x + (size_t)(b0 + bb) * II + js * 4, &xt[0][bb * JT + js * 4]);
  }

  const int NC = II / JT;  // 4
  for (int c = 0; c < NC; ++c) {
    const int cur = c & 1;
    if (c + 1 < NC) {
      const int nxt = cur ^ 1;
#pragma unroll
      for (int r = 0; r < 2; ++r) {
        int seg = tid + BLOCK * r;
        int bb = seg >> 4, js = seg & 15;
        async_copy16(x + (size_t)(b0 + bb) * II + (c + 1) * JT + js * 4,
                     &xt[nxt][bb * JT + js * 4]);
      }
      async_wait_le2();
    } else {
      async_wait_le0();
    }
    __syncthreads();

    const float* smuc = smu   + (size_t)(c * JT) * HH + h;
    const float* ssgc = ssig  + (size_t)(c * JT) * HH + h;
    const float* swc  = sW    + (size_t)(c * JT) * HH + h;
    const float* sec  = serev + (size_t)(c * JT) * HH + h;

#pragma unroll 4
    for (int j = 0; j < JT; ++j) {
      float m  = smuc[(size_t)j * HH];
      float sg = ssgc[(size_t)j * HH];
      float w  = swc[(size_t)j * HH];
      float we = w * sec[(size_t)j * HH];
      float hsg = 0.5f * sg;
#pragma unroll
      for (int k = 0; k < BT; ++k) {
        float xv = xt[cur][(bs + 4 * k) * JT + j];       // wave-uniform broadcast
        float t  = fast_tanh((xv - m) * hsg);            // tanh(0.5*(x-mu)*sig)
        float s  = __builtin_fmaf(t, 0.5f, 0.5f);        // sigmoid
        accn[k]  = __builtin_fmaf(we, s, accn[k]);
        accd[k]  = __builtin_fmaf(w,  s, accd[k]);
      }
    }
    __syncthreads();
  }

#pragma unroll
  for (int k = 0; k < BT; ++k) {
    int b = b0 + bs + 4 * k;
    snum[(size_t)b * HH + h] = accn[k];
    sden[(size_t)b * HH + h] = accd[k];
  }
}

// ---------------- one ODE-solver unfold step --------------------------------
__global__ __launch_bounds__(BLOCK) void ltc_step_kernel(
    const float* __restrict__ vprev,  // B x H
    const float* __restrict__ sigma,  // H x H
    const float* __restrict__ nmsig,  // H x H : -(mu*sigma)
    const float* __restrict__ Wm,     // H x H
    const float* __restrict__ wer,    // H x H : W*erev
    const float* __restrict__ snum,   // B x H
    const float* __restrict__ sden,   // B x H
    const float* __restrict__ vleak,  // H
    const float* __restrict__ gleak,  // H
    const float* __restrict__ cm,     // H
    float* __restrict__ vnext) {      // B x H
  __shared__ float vt[2][TILE_B * JT];

  const int tid = threadIdx.x;
  const int hl  = tid & 63;
  const int bs  = tid >> 6;  // 0..3
  const int h   = blockIdx.x * TILE_H + hl;
  const int b0  = blockIdx.y * TILE_B;

  float accn[BT], accd[BT];
#pragma unroll
  for (int k = 0; k < BT; ++k) { accn[k] = 0.f; accd[k] = 0.f; }

  // issue chunk 0 of the v_pre tile
#pragma unroll
  for (int r = 0; r < 2; ++r) {
    int seg = tid + BLOCK * r;
    int bb = seg >> 4, js = seg & 15;
    async_copy16(vprev + (size_t)(b0 + bb) * HH + js * 4, &vt[0][bb * JT + js * 4]);
  }

  const int NC = HH / JT;  // 8
  for (int c = 0; c < NC; ++c) {
    const int cur = c & 1;
    if (c + 1 < NC) {
      const int nxt = cur ^ 1;
#pragma unroll
      for (int r = 0; r < 2; ++r) {
        int seg = tid + BLOCK * r;
        int bb = seg >> 4, js = seg & 15;
        async_copy16(vprev + (size_t)(b0 + bb) * HH + (c + 1) * JT + js * 4,
                     &vt[nxt][bb * JT + js * 4]);
      }
      async_wait_le2();  // in-order: ensures current chunk's copies landed
    } else {
      async_wait_le0();
    }
    __syncthreads();

    const float* sgc = sigma + (size_t)(c * JT) * HH + h;
    const float* nmc = nmsig + (size_t)(c * JT) * HH + h;
    const float* wc  = Wm    + (size_t)(c * JT) * HH + h;
    const float* wec = wer   + (size_t)(c * JT) * HH + h;

#pragma unroll 4
    for (int j = 0; j < JT; ++j) {
      float sg = sgc[(size_t)j * HH];
      float nm = nmc[(size_t)j * HH];
      float w  = wc[(size_t)j * HH];
      float we = wec[(size_t)j * HH];
      float hsg = 0.5f * sg;   // tanh-arg scaling: sigmoid(a)=0.5*tanh(a/2)+0.5
      float hnm = 0.5f * nm;
#pragma unroll
      for (int k = 0; k < BT; ++k) {
        float v = vt[cur][(bs + 4 * k) * JT + j];        // wave-uniform broadcast
        float t = fast_tanh(__builtin_fmaf(v, hsg, hnm));
        float s = __builtin_fmaf(t, 0.5f, 0.5f);
        accn[k] = __builtin_fmaf(we, s, accn[k]);
        accd[k] = __builtin_fmaf(w,  s, accd[k]);
      }
    }
    __syncthreads();
  }

  const float cmh  = cm[h];
  const float glh  = gleak[h];
  const float glv  = glh * vleak[h];
  const float denb = cmh + glh;
#pragma unroll
  for (int k = 0; k < BT; ++k) {
    int b = b0 + bs + 4 * k;
    float v   = vprev[(size_t)b * HH + h];
    float num = __builtin_fmaf(cmh, v, glv) + snum[(size_t)b * HH + h] + accn[k];
    float den = denb + sden[(size_t)b * HH + h] + accd[k];
    vnext[(size_t)b * HH + h] = num / (den + 1e-8f);
  }
}

// ---------------------------------------------------------------------------
extern "C" void kernel_launch(void* const* d_in, const int* in_sizes, int n_in,
                              void* d_out, int out_size, void* d_ws, size_t ws_size,
                              hipStream_t stream) {
  const float* x     = (const float*)d_in[0];   // inputs  (B,I)
  const float* state = (const float*)d_in[1];   // state   (B,H)
  const float* smu   = (const float*)d_in[2];   // sensory_mu    (I,H)
  const float* ssig  = (const float*)d_in[3];   // sensory_sigma (I,H)
  const float* sW    = (const float*)d_in[4];   // sensory_W     (I,H)
  const float* serev = (const float*)d_in[5];   // sensory_erev  (I,H)
  const float* mu    = (const float*)d_in[6];   // mu    (H,H)
  const float* sigma = (const float*)d_in[7];   // sigma (H,H)
  const float* W     = (const float*)d_in[8];   // W     (H,H)
  const float* erev  = (const float*)d_in[9];   // erev  (H,H)
  const float* vleak = (const float*)d_in[10];  // (H)
  const float* gleak = (const float*)d_in[11];  // (H)
  const float* cm    = (const float*)d_in[12];  // (H)
  float* out = (float*)d_out;

  float* ws    = (float*)d_ws;
  float* nmsig = ws;                        // H*H
  float* wer   = nmsig + (size_t)HH * HH;   // H*H
  float* snum  = wer   + (size_t)HH * HH;   // B*H
  float* sden  = snum  + (size_t)BB * HH;   // B*H
  float* v0    = sden  + (size_t)BB * HH;   // B*H
  float* v1    = v0    + (size_t)BB * HH;   // B*H

  ltc_prep_kernel<<<(HH * HH + 255) / 256, 256, 0, stream>>>(
      mu, sigma, W, erev, nmsig, wer, HH * HH);

  dim3 grid(HH / TILE_H, BB / TILE_B);  // (8,16) = 128 blocks
  ltc_sensory_kernel<<<grid, BLOCK, 0, stream>>>(x, smu, ssig, sW, serev, snum, sden);

  const float* cur = state;
  float* bufs[2] = {v0, v1};
  for (int s = 0; s < UNFOLDS; ++s) {
    float* nxt = (s == UNFOLDS - 1) ? out : bufs[s & 1];
    ltc_step_kernel<<<grid, BLOCK, 0, stream>>>(
        cur, sigma, nmsig, W, wer, snum, sden, vleak, gleak, cm, nxt);
    cur = nxt;
  }
}